// HeteroRGCNLayer_86715389706548
// MI455X (gfx1250) — compile-verified
//
#include <hip/hip_runtime.h>
#include <hip/hip_bf16.h>

// ---------------------------------------------------------------------------
// HeteroRGCN layer for MI455X (gfx1250, wave32)
//   h = x @ ((W0+W1+W2)/3)^T          -> fp32 WMMA 16x16x4
//   a_e = segment_mean(h[src_e], dst_e)
//   out = LayerNorm(ReLU((a0+a1+a2)/3)) * gamma + beta
// The three segment means are folded into one weighted atomic scatter using
// per-node weights 1/(3*max(cnt_e,1)), accumulated directly into d_out.
// GEMM blocks cover 32 rows so each B fragment feeds two WMMAs (2x reuse).
// ---------------------------------------------------------------------------

#define N_NODES 100000
#define D_IN    512
#define D_OUT   512
#define N_EDGE  200000
#define LN_EPS  1e-5f

typedef __attribute__((ext_vector_type(2))) float v2f;
typedef __attribute__((ext_vector_type(8))) float v8f;

#define LDS_STRIDE 516   // 516 % 64 == 4 -> conflict-free ds_load_b64 A-fragments

// ---------------------------------------------------------------------------
// 0) zero a buffer, 16B at a time
// ---------------------------------------------------------------------------
__global__ void zero4_kernel(float4* __restrict__ p, long long n4) {
    long long i = (long long)blockIdx.x * blockDim.x + threadIdx.x;
    if (i < n4) p[i] = make_float4(0.f, 0.f, 0.f, 0.f);
}

// ---------------------------------------------------------------------------
// 1) fold (W0+W1+W2)/3 and pack transposed into WMMA-B fragment order.
//    B tile (4x16) for output-col tile tg and k-step s, lane l:
//      col  = tg*16 + (l & 15)
//      krow = s*4  + (l >> 4)*2
//      frag = { Wsum[col][krow], Wsum[col][krow+1] }   (VGPR0, VGPR1)
//    Stored at packedB[((tg*128 + s)*32 + l)]  (float2 units)
// ---------------------------------------------------------------------------
__global__ void pack_w_kernel(const float* __restrict__ W0,
                              const float* __restrict__ W1,
                              const float* __restrict__ W2,
                              float2* __restrict__ packedB) {
    int idx = blockIdx.x * blockDim.x + threadIdx.x;   // 32*128*32 = 131072
    if (idx >= 32 * 128 * 32) return;
    int lane = idx & 31;
    int s    = (idx >> 5) & 127;
    int tg   = idx >> 12;
    int col  = tg * 16 + (lane & 15);
    int krow = s * 4 + (lane >> 4) * 2;
    const float inv3 = 1.0f / 3.0f;
    long long o0 = (long long)col * D_IN + krow;
    float b0 = (W0[o0]     + W1[o0]     + W2[o0])     * inv3;
    float b1 = (W0[o0 + 1] + W1[o0 + 1] + W2[o0 + 1]) * inv3;
    packedB[idx] = make_float2(b0, b1);
}

// ---------------------------------------------------------------------------
// 2) per-etype in-degree counting
// ---------------------------------------------------------------------------
__global__ void count_kernel(const int* __restrict__ dst, int* __restrict__ cnt, int e) {
    int i = blockIdx.x * blockDim.x + threadIdx.x;
    if (i < e) atomicAdd(&cnt[dst[i]], 1);
}

__global__ void invw_kernel(const int* __restrict__ cnt, float* __restrict__ invw, int n) {
    int i = blockIdx.x * blockDim.x + threadIdx.x;
    if (i < n) {
        int c = cnt[i];
        invw[i] = 1.0f / (3.0f * (float)(c > 1 ? c : 1));
    }
}

// ---------------------------------------------------------------------------
// 3) GEMM: h[N,512] = x[N,512] @ Wsum^T   via V_WMMA_F32_16X16X4_F32
//    128 threads (4 waves) per block; block owns 32 rows x all 512 cols.
//    Wave w owns cols [w*128, w*128+128) = 8 col-tiles x 2 row-tiles.
//    Each B fragment is reused by two WMMAs -> 16 WMMA : 8 vmem : 2 ds.
// ---------------------------------------------------------------------------
__global__ __launch_bounds__(128)
void gemm_wmma_kernel(const float* __restrict__ x,
                      const float* __restrict__ packedB,
                      float* __restrict__ h) {
    __shared__ float tileA[32 * LDS_STRIDE];

    const int m0  = blockIdx.x * 32;
    const int tid = threadIdx.x;

    // stage 32x512 x-tile into LDS (coalesced float4 global loads)
#pragma unroll
    for (int r = 0; r < 32; ++r) {
        int c = tid * 4;  // 128 threads * 4 floats = 512 = one row
        float4 v = *(const float4*)(x + (long long)(m0 + r) * D_IN + c);
        *(float4*)(&tileA[r * LDS_STRIDE + c]) = v;
    }
    __syncthreads();

    const int wave  = tid >> 5;
    const int lane  = tid & 31;
    const int lrow  = lane & 15;
    const int khalf = (lane >> 4) * 2;      // lanes 16-31 take K=+2,+3
    const int tg0   = wave * 8;             // first global col-tile of this wave

    v8f acc0[8] = {};                        // rows m0 .. m0+15
    v8f acc1[8] = {};                        // rows m0+16 .. m0+31
    const v2f* __restrict__ B = (const v2f*)packedB;

    const int aoff0 = lrow * LDS_STRIDE + khalf;
    const int aoff1 = (16 + lrow) * LDS_STRIDE + khalf;

    for (int s = 0; s < 128; ++s) {
        const int k0 = s * 4;
        // A fragments (16x4 fp32): ds_load_b64, conflict-free with stride 516
        v2f a0 = *(const v2f*)(&tileA[aoff0 + k0]);
        v2f a1 = *(const v2f*)(&tileA[aoff1 + k0]);
#pragma unroll
        for (int t = 0; t < 8; ++t) {
            v2f b = B[((tg0 + t) * 128 + s) * 32 + lane];
            acc0[t] = __builtin_amdgcn_wmma_f32_16x16x4_f32(
                false, a0, false, b, (short)0, acc0[t], false, false);
            acc1[t] = __builtin_amdgcn_wmma_f32_16x16x4_f32(
                false, a1, false, b, (short)0, acc1[t], false, false);
        }
    }

    // C/D layout: VGPR j -> row base + j + (lane>>4)*8, col = cbase + t*16
    const int rbase = m0 + (lane >> 4) * 8;
    const int cbase = wave * 128 + (lane & 15);
#pragma unroll
    for (int t = 0; t < 8; ++t) {
#pragma unroll
        for (int j = 0; j < 8; ++j) {
            h[(long long)(rbase + j) * D_OUT + cbase + t * 16]      = acc0[t][j];
            h[(long long)(rbase + 16 + j) * D_OUT + cbase + t * 16] = acc1[t][j];
        }
    }
}

// ---------------------------------------------------------------------------
// 4) weighted scatter-add: out[dst] += h[src] * invw[dst]
//    256 threads = 2 edges/block, 128 threads/edge, 4 floats/thread
// ---------------------------------------------------------------------------
__global__ __launch_bounds__(256)
void scatter_kernel(const int* __restrict__ src, const int* __restrict__ dst,
                    const float* __restrict__ h, const float* __restrict__ invw,
                    float* __restrict__ out, int e) {
    int slot = blockIdx.x * 2 + (threadIdx.x >> 7);
    if (slot >= e) return;
    int t = threadIdx.x & 127;
    int sN = src[slot];
    int dN = dst[slot];
    float w = invw[dN];
    float4 v = *(const float4*)(h + (long long)sN * D_OUT + t * 4);
    float* o = out + (long long)dN * D_OUT + t * 4;
    atomicAdd(o + 0, v.x * w);
    atomicAdd(o + 1, v.y * w);
    atomicAdd(o + 2, v.z * w);
    atomicAdd(o + 3, v.w * w);
}

// ---------------------------------------------------------------------------
// 5) fused ReLU + LayerNorm, in place. One wave32 per row; 16 floats/lane.
// ---------------------------------------------------------------------------
__global__ __launch_bounds__(256)
void relu_layernorm_kernel(float* __restrict__ out,
                           const float* __restrict__ gamma,
                           const float* __restrict__ beta, int n) {
    int row = blockIdx.x * 8 + (threadIdx.x >> 5);
    if (row >= n) return;
    int lane = threadIdx.x & 31;
    float* p = out + (long long)row * D_OUT;

    float4 v[4];
    float s = 0.f, sq = 0.f;
#pragma unroll
    for (int j = 0; j < 4; ++j) {
        int c = j * 128 + lane * 4;
        v[j] = *(const float4*)(p + c);
        v[j].x = fmaxf(v[j].x, 0.f);
        v[j].y = fmaxf(v[j].y, 0.f);
        v[j].z = fmaxf(v[j].z, 0.f);
        v[j].w = fmaxf(v[j].w, 0.f);
        s  += v[j].x + v[j].y + v[j].z + v[j].w;
        sq += v[j].x * v[j].x + v[j].y * v[j].y + v[j].z * v[j].z + v[j].w * v[j].w;
    }
#pragma unroll
    for (int m = 16; m >= 1; m >>= 1) {
        s  += __shfl_xor(s,  m, 32);
        sq += __shfl_xor(sq, m, 32);
    }
    const float inv_d = 1.0f / (float)D_OUT;
    float mu  = s * inv_d;
    float var = sq * inv_d - mu * mu;
    float rs  = rsqrtf(var + LN_EPS);
#pragma unroll
    for (int j = 0; j < 4; ++j) {
        int c = j * 128 + lane * 4;
        float4 g = *(const float4*)(gamma + c);
        float4 b = *(const float4*)(beta + c);
        float4 o;
        o.x = (v[j].x - mu) * rs * g.x + b.x;
        o.y = (v[j].y - mu) * rs * g.y + b.y;
        o.z = (v[j].z - mu) * rs * g.z + b.z;
        o.w = (v[j].w - mu) * rs * g.w + b.w;
        *(float4*)(p + c) = o;
    }
}

// ---------------------------------------------------------------------------
// launch
// ---------------------------------------------------------------------------
extern "C" void kernel_launch(void* const* d_in, const int* in_sizes, int n_in,
                              void* d_out, int out_size, void* d_ws, size_t ws_size,
                              hipStream_t stream) {
    const float* x     = (const float*)d_in[0];
    const float* W0    = (const float*)d_in[1];
    const float* W1    = (const float*)d_in[2];
    const float* W2    = (const float*)d_in[3];
    const float* gamma = (const float*)d_in[4];
    const float* beta  = (const float*)d_in[5];
    const int*   src0  = (const int*)d_in[6];
    const int*   dst0  = (const int*)d_in[7];
    const int*   src1  = (const int*)d_in[8];
    const int*   dst1  = (const int*)d_in[9];
    const int*   src2  = (const int*)d_in[10];
    const int*   dst2  = (const int*)d_in[11];
    float* out = (float*)d_out;

    // workspace layout
    char* ws = (char*)d_ws;
    const size_t h_bytes  = (size_t)N_NODES * D_OUT * sizeof(float);   // 204,800,000
    const size_t pb_bytes = (size_t)32 * 128 * 32 * sizeof(float2);    //   1,048,576
    const size_t ct_bytes = (size_t)3 * N_NODES * sizeof(int);         //   1,200,000
    float*  h       = (float*)ws;
    float2* packedB = (float2*)(ws + h_bytes);
    int*    cnt     = (int*)(ws + h_bytes + pb_bytes);
    float*  invw    = (float*)(ws + h_bytes + pb_bytes + ct_bytes);
    (void)ws_size; (void)in_sizes; (void)n_in; (void)out_size;

    // 0) zero accumulator (d_out) and counts
    {
        long long n4 = (long long)N_NODES * D_OUT / 4;                 // 12.8M
        zero4_kernel<<<(unsigned)((n4 + 255) / 256), 256, 0, stream>>>((float4*)out, n4);
        long long c4 = (long long)3 * N_NODES / 4;                     // 75000
        zero4_kernel<<<(unsigned)((c4 + 255) / 256), 256, 0, stream>>>((float4*)cnt, c4);
    }

    // 1) fold + pack weights
    pack_w_kernel<<<512, 256, 0, stream>>>(W0, W1, W2, packedB);

    // 2) in-degree counts + weights
    const int cblocks = (N_EDGE + 255) / 256;
    count_kernel<<<cblocks, 256, 0, stream>>>(dst0, cnt + 0 * N_NODES, N_EDGE);
    count_kernel<<<cblocks, 256, 0, stream>>>(dst1, cnt + 1 * N_NODES, N_EDGE);
    count_kernel<<<cblocks, 256, 0, stream>>>(dst2, cnt + 2 * N_NODES, N_EDGE);
    invw_kernel<<<(3 * N_NODES + 255) / 256, 256, 0, stream>>>(cnt, invw, 3 * N_NODES);

    // 3) WMMA GEMM (32 rows per block)
    gemm_wmma_kernel<<<N_NODES / 32, 128, 0, stream>>>(x, (const float*)packedB, h);

    // 4) weighted scatter into d_out
    scatter_kernel<<<N_EDGE / 2, 256, 0, stream>>>(src0, dst0, h, invw + 0 * N_NODES, out, N_EDGE);
    scatter_kernel<<<N_EDGE / 2, 256, 0, stream>>>(src1, dst1, h, invw + 1 * N_NODES, out, N_EDGE);
    scatter_kernel<<<N_EDGE / 2, 256, 0, stream>>>(src2, dst2, h, invw + 2 * N_NODES, out, N_EDGE);

    // 5) ReLU + LayerNorm in place
    relu_layernorm_kernel<<<(N_NODES + 7) / 8, 256, 0, stream>>>(out, gamma, beta, N_NODES);
}